// GATModel_90649579750147
// MI455X (gfx1250) — compile-verified
//
#include <hip/hip_runtime.h>
#include <math.h>

// Problem constants (match reference)
#define N_NODES 50000          // == 3125 * 16, multiple of 16
#define N_EDGES 800000
#define F_IN    86
#define F_PAD   88             // F_IN padded to multiple of 4 for WMMA K-loop
#define H_DIM   128
#define ED_DIM  16
#define C_CLS   18
#define G_GRAPHS 64
#define NEG_SLOPE 0.2f

typedef __attribute__((ext_vector_type(2)))  float v2f;
typedef __attribute__((ext_vector_type(8)))  float v8f;

// ---------------------------------------------------------------------------
// float atomic max via signed-int max (positive) / unsigned-int min (negative)
// ---------------------------------------------------------------------------
__device__ __forceinline__ void atomicMaxF(float* addr, float v) {
    if (v >= 0.0f) {
        atomicMax((int*)addr, __float_as_int(v));
    } else {
        atomicMin((unsigned int*)addr, __float_as_uint(v));
    }
}

// ---------------------------------------------------------------------------
// Pad-copy: out[i, 0:K] = in[i, 0:K], out[i, K:Kpad] = 0   (row-major)
// ---------------------------------------------------------------------------
__global__ void pad_rows(const float* __restrict__ in, float* __restrict__ out,
                         int M, int K, int Kpad)
{
    const int idx = blockIdx.x * blockDim.x + threadIdx.x;
    if (idx >= M * Kpad) return;
    const int r = idx / Kpad;
    const int k = idx - r * Kpad;
    out[idx] = (k < K) ? in[r * K + k] : 0.0f;
}

// ---------------------------------------------------------------------------
// Transpose + pad: Wt[c, k] = (k < K) ? W[k, c] : 0    W:[K,Nc] -> Wt:[Nc,Kpad]
// ---------------------------------------------------------------------------
__global__ void transpose_pad(const float* __restrict__ W, float* __restrict__ Wt,
                              int K, int Kpad, int Nc)
{
    const int idx = blockIdx.x * blockDim.x + threadIdx.x;
    if (idx >= Nc * Kpad) return;
    const int c = idx / Kpad;
    const int k = idx - c * Kpad;
    Wt[idx] = (k < K) ? W[k * Nc + c] : 0.0f;
}

// ---------------------------------------------------------------------------
// WMMA GEMM: D[M,Nc] = A[M,K] * Bt[Nc,K]^T   (fp32, V_WMMA_F32_16X16X4_F32)
// Preconditions: M % 16 == 0, K % 4 == 0, Nc % 16 == 0. One wave per 16x16 tile.
// A layout per ISA: lane L -> row = L&15, holds K = 2*(L>>4)+{0,1}  (float2)
// B layout mirrored; D: lane L, vgpr r -> row = 8*(L>>4)+r, col = L&15.
// ---------------------------------------------------------------------------
__global__ void gemm_wmma(const float* __restrict__ A,
                          const float* __restrict__ Bt,
                          float* __restrict__ D,
                          int M, int K, int Nc)
{
    const int lane = threadIdx.x & 31;
    const int wave = threadIdx.x >> 5;
    const int wpb  = blockDim.x >> 5;
    const int tilesN = Nc >> 4;
    const int tilesM = M >> 4;
    const int tile = blockIdx.x * wpb + wave;
    if (tile >= tilesM * tilesN) return;

    const int tm   = tile / tilesN;
    const int tn   = tile - tm * tilesN;
    const int row0 = tm << 4;
    const int col0 = tn << 4;

    const int mr  = lane & 15;
    const int hi  = lane >> 4;
    const int ra  = row0 + mr;      // A row this lane reads
    const int col = col0 + mr;      // B/D column this lane handles

    const float2* __restrict__ Ap = (const float2*)(A  + (size_t)ra  * K);
    const float2* __restrict__ Bp = (const float2*)(Bt + (size_t)col * K);

    v8f acc = {};
    int idx = hi;                   // float2 index: (k + 2*hi)/2
#pragma unroll 8
    for (int k = 0; k < K; k += 4, idx += 2) {
        const float2 a = Ap[idx];
        const float2 b = Bp[idx];
        v2f av; av.x = a.x; av.y = a.y;
        v2f bv; bv.x = b.x; bv.y = b.y;
        acc = __builtin_amdgcn_wmma_f32_16x16x4_f32(
            false, av, false, bv, (short)0, acc, false, false);
    }

    const int mbase = row0 + (hi << 3);
#pragma unroll
    for (int r = 0; r < 8; ++r)
        D[(size_t)(mbase + r) * Nc + col] = acc[r];
}

// ---------------------------------------------------------------------------
// s[i] = h[i,:] . a_src ; d[i] = h[i,:] . a_dst   (one wave per node)
// ---------------------------------------------------------------------------
__global__ void node_dots(const float* __restrict__ h,
                          const float* __restrict__ a_src,
                          const float* __restrict__ a_dst,
                          float* __restrict__ s, float* __restrict__ d)
{
    const int lane = threadIdx.x & 31;
    const int node = (blockIdx.x * blockDim.x + threadIdx.x) >> 5;
    if (node >= N_NODES) return;
    const float4* hp = (const float4*)(h + (size_t)node * H_DIM);
    const float4* asp = (const float4*)a_src;
    const float4* adp = (const float4*)a_dst;
    const float4 hv = hp[lane];
    const float4 av = asp[lane];
    const float4 dv = adp[lane];
    float ss = hv.x * av.x + hv.y * av.y + hv.z * av.z + hv.w * av.w;
    float dd = hv.x * dv.x + hv.y * dv.y + hv.z * dv.z + hv.w * dv.w;
#pragma unroll
    for (int off = 16; off > 0; off >>= 1) {
        ss += __shfl_down(ss, off, 32);
        dd += __shfl_down(dd, off, 32);
    }
    if (lane == 0) { s[node] = ss; d[node] = dd; }
}

// wae[j] = We[j,:] . a_edge   (ED_DIM = 16 outputs; tiny)
__global__ void compute_wae(const float* __restrict__ We,
                            const float* __restrict__ ae,
                            float* __restrict__ wae)
{
    const int j = threadIdx.x;
    if (j < ED_DIM) {
        float acc = 0.0f;
        for (int c = 0; c < H_DIM; ++c) acc += We[j * H_DIM + c] * ae[c];
        wae[j] = acc;
    }
}

// zero agg[N*H]; m = -inf, denom = 0 for first N threads
__global__ void init_layer(float* __restrict__ agg,
                           float* __restrict__ m,
                           float* __restrict__ denom)
{
    const int idx = blockIdx.x * blockDim.x + threadIdx.x;
    if (idx < N_NODES * H_DIM) agg[idx] = 0.0f;
    if (idx < N_NODES) { m[idx] = -INFINITY; denom[idx] = 0.0f; }
}

// alpha = leaky_relu(s[src]+d[dst]+<eattr,wae>), store + segment max into m[dst]
__global__ void edge_alpha_max(const int* __restrict__ src,
                               const int* __restrict__ dst,
                               const float* __restrict__ eattr,
                               const float* __restrict__ wae,
                               const float* __restrict__ s,
                               const float* __restrict__ d,
                               float* __restrict__ alpha,
                               float* __restrict__ m)
{
    const int e = blockIdx.x * blockDim.x + threadIdx.x;
    if (e >= N_EDGES) return;
    const float4* ep = (const float4*)(eattr + (size_t)e * ED_DIM);
    const float4* wp = (const float4*)wae;
    float acc = 0.0f;
#pragma unroll
    for (int q = 0; q < 4; ++q) {
        const float4 a = ep[q], b = wp[q];
        acc += a.x * b.x + a.y * b.y + a.z * b.z + a.w * b.w;
    }
    float al = s[src[e]] + d[dst[e]] + acc;
    al = (al > 0.0f) ? al : NEG_SLOPE * al;
    alpha[e] = al;
    atomicMaxF(&m[dst[e]], al);
}

// ex = exp(alpha - m[dst]) (in place); denom[dst] += ex
__global__ void edge_exp_sum(const int* __restrict__ dst,
                             const float* __restrict__ m,
                             float* __restrict__ alpha_ex,
                             float* __restrict__ denom)
{
    const int e = blockIdx.x * blockDim.x + threadIdx.x;
    if (e >= N_EDGES) return;
    const int t = dst[e];
    const float ex = expf(alpha_ex[e] - m[t]);
    alpha_ex[e] = ex;
    atomicAdd(&denom[t], ex);
}

// agg[dst,:] += h[src,:] * (ex / max(denom[dst],1e-16))   (one wave per edge)
__global__ void edge_aggregate(const int* __restrict__ src,
                               const int* __restrict__ dst,
                               const float* __restrict__ ex,
                               const float* __restrict__ denom,
                               const float* __restrict__ h,
                               float* __restrict__ agg)
{
    const int lane = threadIdx.x & 31;
    const int e = (blockIdx.x * blockDim.x + threadIdx.x) >> 5;
    if (e >= N_EDGES) return;
    const int sn = src[e], tn = dst[e];
    const float w = ex[e] / fmaxf(denom[tn], 1e-16f);
    const float4 hv = ((const float4*)(h + (size_t)sn * H_DIM))[lane];
    float* op = agg + (size_t)tn * H_DIM + lane * 4;
    atomicAdd(op + 0, hv.x * w);
    atomicAdd(op + 1, hv.y * w);
    atomicAdd(op + 2, hv.z * w);
    atomicAdd(op + 3, hv.w * w);
}

// agg = relu(agg + bias) in place
__global__ void bias_relu(float* __restrict__ agg, const float* __restrict__ bias)
{
    const int idx = blockIdx.x * blockDim.x + threadIdx.x;
    if (idx >= N_NODES * H_DIM) return;
    const float v = agg[idx] + bias[idx & (H_DIM - 1)];
    agg[idx] = fmaxf(v, 0.0f);
}

__global__ void pool_init(float* __restrict__ pooled)
{
    const int idx = blockIdx.x * blockDim.x + threadIdx.x;
    if (idx < G_GRAPHS * H_DIM) pooled[idx] = -INFINITY;
}

__global__ void pool_max(const float* __restrict__ h,
                         const int* __restrict__ batch,
                         float* __restrict__ pooled)
{
    const int idx = blockIdx.x * blockDim.x + threadIdx.x;
    if (idx >= N_NODES * H_DIM) return;
    const int node = idx >> 7;          // H_DIM = 128
    const int c = idx & (H_DIM - 1);
    atomicMaxF(&pooled[batch[node] * H_DIM + c], h[idx]);
}

// logits = pooled(finite-guarded) @ Wl + bl ; out = log_softmax(logits)
__global__ void classifier(const float* __restrict__ pooled,
                           const float* __restrict__ Wl,
                           const float* __restrict__ bl,
                           float* __restrict__ out)
{
    const int g = blockIdx.x * blockDim.x + threadIdx.x;
    if (g >= G_GRAPHS) return;
    float logit[C_CLS];
    for (int c = 0; c < C_CLS; ++c) logit[c] = bl[c];
    for (int hh = 0; hh < H_DIM; ++hh) {
        float p = pooled[g * H_DIM + hh];
        if (!isfinite(p)) p = 0.0f;
        for (int c = 0; c < C_CLS; ++c) logit[c] += p * Wl[hh * C_CLS + c];
    }
    float mx = logit[0];
    for (int c = 1; c < C_CLS; ++c) mx = fmaxf(mx, logit[c]);
    float se = 0.0f;
    for (int c = 0; c < C_CLS; ++c) se += expf(logit[c] - mx);
    const float lse = logf(se) + mx;
    for (int c = 0; c < C_CLS; ++c) out[g * C_CLS + c] = logit[c] - lse;
}

// ---------------------------------------------------------------------------
extern "C" void kernel_launch(void* const* d_in, const int* in_sizes, int n_in,
                              void* d_out, int out_size, void* d_ws, size_t ws_size,
                              hipStream_t stream)
{
    const float* x     = (const float*)d_in[0];
    const int*   ei    = (const int*)  d_in[1];
    const float* eattr = (const float*)d_in[2];
    const int*   batch = (const int*)  d_in[3];
    const float* W1  = (const float*)d_in[4];
    const float* We1 = (const float*)d_in[5];
    const float* as1 = (const float*)d_in[6];
    const float* ad1 = (const float*)d_in[7];
    const float* ae1 = (const float*)d_in[8];
    const float* b1  = (const float*)d_in[9];
    const float* W2  = (const float*)d_in[10];
    const float* We2 = (const float*)d_in[11];
    const float* as2 = (const float*)d_in[12];
    const float* ad2 = (const float*)d_in[13];
    const float* ae2 = (const float*)d_in[14];
    const float* b2  = (const float*)d_in[15];
    const float* Wl  = (const float*)d_in[16];
    const float* bl  = (const float*)d_in[17];

    const int* srcI = ei;               // edge_index[0,:]
    const int* dstI = ei + N_EDGES;     // edge_index[1,:]

    // workspace layout (floats)
    float* ws = (float*)d_ws;
    float* hmat   = ws;  ws += (size_t)N_NODES * H_DIM;
    float* agg    = ws;  ws += (size_t)N_NODES * H_DIM;
    float* xpad   = ws;  ws += (size_t)N_NODES * F_PAD;
    float* w1t    = ws;  ws += H_DIM * F_PAD;     // W1^T zero-padded: [H][F_PAD]
    float* w2t    = ws;  ws += H_DIM * H_DIM;     // W2^T: [H][H]
    float* sArr   = ws;  ws += N_NODES;
    float* dArr   = ws;  ws += N_NODES;
    float* mArr   = ws;  ws += N_NODES;
    float* denArr = ws;  ws += N_NODES;
    float* exArr  = ws;  ws += N_EDGES;
    float* waeArr = ws;  ws += ED_DIM;
    float* pooled = ws;  ws += G_GRAPHS * H_DIM;

    const int TB = 256;
    const int tilesTot = (N_NODES / 16) * (H_DIM / 16);
    const int gemmBlocks = (tilesTot + (TB / 32) - 1) / (TB / 32);
    const int nodeWaveBlocks = (N_NODES * 32 + TB - 1) / TB;
    const int nhBlocks = (N_NODES * H_DIM + TB - 1) / TB;
    const int eBlocks  = (N_EDGES + TB - 1) / TB;
    const int eWaveBlocks = (int)(((size_t)N_EDGES * 32 + TB - 1) / TB);

    // ---------------- Prep: pad x, transpose weights ----------------
    pad_rows<<<(N_NODES * F_PAD + TB - 1) / TB, TB, 0, stream>>>(x, xpad, N_NODES, F_IN, F_PAD);
    transpose_pad<<<(H_DIM * F_PAD + TB - 1) / TB, TB, 0, stream>>>(W1, w1t, F_IN, F_PAD, H_DIM);
    transpose_pad<<<(H_DIM * H_DIM + TB - 1) / TB, TB, 0, stream>>>(W2, w2t, H_DIM, H_DIM, H_DIM);

    // ---------------- Layer 1 ----------------
    gemm_wmma<<<gemmBlocks, TB, 0, stream>>>(xpad, w1t, hmat, N_NODES, F_PAD, H_DIM);
    node_dots<<<nodeWaveBlocks, TB, 0, stream>>>(hmat, as1, ad1, sArr, dArr);
    compute_wae<<<1, 64, 0, stream>>>(We1, ae1, waeArr);
    init_layer<<<nhBlocks, TB, 0, stream>>>(agg, mArr, denArr);
    edge_alpha_max<<<eBlocks, TB, 0, stream>>>(srcI, dstI, eattr, waeArr, sArr, dArr, exArr, mArr);
    edge_exp_sum<<<eBlocks, TB, 0, stream>>>(dstI, mArr, exArr, denArr);
    edge_aggregate<<<eWaveBlocks, TB, 0, stream>>>(srcI, dstI, exArr, denArr, hmat, agg);
    bias_relu<<<nhBlocks, TB, 0, stream>>>(agg, b1);

    // ---------------- Layer 2 ---------------- (gemm consumes agg before re-zero)
    gemm_wmma<<<gemmBlocks, TB, 0, stream>>>(agg, w2t, hmat, N_NODES, H_DIM, H_DIM);
    node_dots<<<nodeWaveBlocks, TB, 0, stream>>>(hmat, as2, ad2, sArr, dArr);
    compute_wae<<<1, 64, 0, stream>>>(We2, ae2, waeArr);
    init_layer<<<nhBlocks, TB, 0, stream>>>(agg, mArr, denArr);
    edge_alpha_max<<<eBlocks, TB, 0, stream>>>(srcI, dstI, eattr, waeArr, sArr, dArr, exArr, mArr);
    edge_exp_sum<<<eBlocks, TB, 0, stream>>>(dstI, mArr, exArr, denArr);
    edge_aggregate<<<eWaveBlocks, TB, 0, stream>>>(srcI, dstI, exArr, denArr, hmat, agg);
    bias_relu<<<nhBlocks, TB, 0, stream>>>(agg, b2);

    // ---------------- Pool + classify ----------------
    pool_init<<<(G_GRAPHS * H_DIM + TB - 1) / TB, TB, 0, stream>>>(pooled);
    pool_max<<<nhBlocks, TB, 0, stream>>>(agg, batch, pooled);
    classifier<<<1, 64, 0, stream>>>(pooled, Wl, bl, (float*)d_out);
}